// HyboNet_22136261444134
// MI455X (gfx1250) — compile-verified
//
#include <hip/hip_runtime.h>

#define LEPS 1e-8f

typedef __attribute__((ext_vector_type(2))) float v2f;
typedef __attribute__((ext_vector_type(8))) float v8f;

__device__ __forceinline__ float reduce16(float v) {
  // sum within each 16-lane half of the wave32
  v += __shfl_xor(v, 1, 32);
  v += __shfl_xor(v, 2, 32);
  v += __shfl_xor(v, 4, 32);
  v += __shfl_xor(v, 8, 32);
  return v;
}
__device__ __forceinline__ float reduce32(float v) {
  v += __shfl_xor(v, 1, 32);
  v += __shfl_xor(v, 2, 32);
  v += __shfl_xor(v, 4, 32);
  v += __shfl_xor(v, 8, 32);
  v += __shfl_xor(v, 16, 32);
  return v;
}

// ---------------- exp map: x (N x 63) -> p (N x 64), one wave per node ----------------
__global__ void expmap_kernel(const float* __restrict__ x, float* __restrict__ p, int n) {
  int wave = threadIdx.x >> 5;
  int lane = threadIdx.x & 31;
  int node = blockIdx.x * 8 + wave;
  if (node >= n) return;
  const float* u = x + (size_t)node * 63;
  int i0 = 2 * lane, i1 = i0 + 1;
  float a = (i0 < 63) ? u[i0] : 0.f;
  float b = (i1 < 63) ? u[i1] : 0.f;
  float ss = reduce32(a * a + b * b);
  float nrm = fmaxf(sqrtf(ss), LEPS);
  float f = sinhf(nrm) / nrm;
  float ptime = sqrtf(1.f + f * f * ss);
  float* pr = p + (size_t)node * 64;
  if (lane == 0) pr[0] = ptime;
  if (i0 < 63) pr[1 + i0] = f * a;
  if (i1 < 63) pr[1 + i1] = f * b;
}

// ------------- Lorentz linear: H = act(X @ W^T + b), fp32 WMMA, fused epilogue -------------
// One wave computes a 16x64 output tile: 4 accumulators, K-loop of 16 x (16x16x4) WMMAs.
__global__ void lorentz_linear_kernel(const float* __restrict__ X,
                                      const float* __restrict__ W,
                                      const float* __restrict__ bias,
                                      const float* __restrict__ log_scale,
                                      float* __restrict__ H,
                                      int n, int do_relu) {
  int wave = threadIdx.x >> 5;
  int lane = threadIdx.x & 31;
  int row0 = (blockIdx.x * 8 + wave) * 16;
  if (row0 >= n) return;                 // wave-uniform: EXEC stays all-1s for WMMA waves
  int half = lane >> 4;                  // 0: K=k0,k0+1 / rows 0-7 ; 1: K=k0+2,k0+3 / rows 8-15
  int ln = lane & 15;

  v8f acc[4];
#pragma unroll
  for (int t = 0; t < 4; ++t) {
    float bv = bias[ln + 16 * t];        // C(m, col) = b[col] for all m
    v8f tmp = {bv, bv, bv, bv, bv, bv, bv, bv};
    acc[t] = tmp;
  }

  int arow = row0 + ln;
  if (arow >= n) arow = n - 1;           // clamp reads; padded rows never stored
  const float* xr = X + (size_t)arow * 64;

#pragma unroll
  for (int k0 = 0; k0 < 64; k0 += 4) {
    int kb = k0 + half * 2;
    v2f a;
    a.x = xr[kb];
    a.y = xr[kb + 1];
    if (do_relu) { a.x = fmaxf(a.x, 0.f); a.y = fmaxf(a.y, 0.f); }
#pragma unroll
    for (int t = 0; t < 4; ++t) {
      int col = ln + 16 * t;             // B(k, col) = W[col*64 + k]  (h = x @ W^T)
      v2f bb;
      bb.x = W[col * 64 + kb];
      bb.y = W[col * 64 + kb + 1];
      acc[t] = __builtin_amdgcn_wmma_f32_16x16x4_f32(
          false, a, false, bb, (short)0, acc[t], false, false);
    }
  }

  // Epilogue: time = sigmoid(h0)*exp(s)+1.1 ; space *= sqrt((time^2-1)/max(|space|^2, eps))
  float ls = __expf(log_scale[0]);
#pragma unroll
  for (int r = 0; r < 8; ++r) {
    float partial = 0.f;
#pragma unroll
    for (int t = 0; t < 4; ++t) {
      float v = acc[t][r];
      partial += (t == 0 && ln == 0) ? 0.f : v * v;   // exclude time column
    }
    partial = reduce16(partial);                       // per-row |space|^2
    float h0 = __shfl(acc[0][r], lane & 16, 32);       // col 0 lives in lane 0 / lane 16
    float time = ls / (1.f + __expf(-h0)) + 1.1f;
    float sq = fmaxf(partial, LEPS);
    float scl = sqrtf((time * time - 1.f) / sq);
    int m = row0 + r + 8 * half;
    if (m < n) {
      float* hr = H + (size_t)m * 64;
#pragma unroll
      for (int t = 0; t < 4; ++t) {
        int col = ln + 16 * t;
        hr[col] = (col == 0) ? time : acc[t][r] * scl;
      }
    }
  }
}

// ---------------- accumulator zero ----------------
__global__ void zero_kernel(float* __restrict__ s, int n) {
  int i = blockIdx.x * 256 + threadIdx.x;
  if (i < n) s[i] = 0.f;
}

// ---------------- edge scatter: S[dst] += w * H[src]; 64 lanes per edge ----------------
__global__ void scatter_kernel(const float* __restrict__ H, const int* __restrict__ src,
                               const int* __restrict__ dst, const float* __restrict__ w,
                               float* __restrict__ S, unsigned e_count) {
  unsigned t = blockIdx.x * 256u + threadIdx.x;
  unsigned e = t >> 6;
  if (e >= e_count) return;
  int c = (int)(t & 63u);
  float val = w[e] * H[(size_t)src[e] * 64 + c];
  atomicAdd(&S[(size_t)dst[e] * 64 + c], val);       // L2-resident (20.5 MB << 192 MB)
}

// ---------------- to_manifold: O = S / sqrt(max(|<S,S>_L|, eps)); one wave per node ----------------
__global__ void normalize_kernel(const float* __restrict__ S, float* __restrict__ O, int n) {
  int wave = threadIdx.x >> 5;
  int lane = threadIdx.x & 31;
  int node = blockIdx.x * 8 + wave;
  if (node >= n) return;
  const float* sr = S + (size_t)node * 64;
  float a = sr[2 * lane], b = sr[2 * lane + 1];
  float ss = reduce32(a * a + b * b);
  float s0 = __shfl(a, 0, 32);
  float inner = ss - 2.f * s0 * s0;                  // -s0^2 + rest
  float inv = 1.f / sqrtf(fmaxf(fabsf(inner), LEPS));
  float* orow = O + (size_t)node * 64;
  orow[2 * lane] = a * inv;
  orow[2 * lane + 1] = b * inv;
}

// ---------------- final: out = to_manifold(h1 + to_manifold(acc2)) ----------------
__global__ void final_kernel(const float* __restrict__ S2, const float* __restrict__ H1,
                             float* __restrict__ OUT, int n) {
  int wave = threadIdx.x >> 5;
  int lane = threadIdx.x & 31;
  int node = blockIdx.x * 8 + wave;
  if (node >= n) return;
  const float* sr = S2 + (size_t)node * 64;
  const float* hr = H1 + (size_t)node * 64;
  float a = sr[2 * lane], b = sr[2 * lane + 1];
  float ss = reduce32(a * a + b * b);
  float s0 = __shfl(a, 0, 32);
  float inv = 1.f / sqrtf(fmaxf(fabsf(ss - 2.f * s0 * s0), LEPS));
  float ua = hr[2 * lane] + a * inv;
  float ub = hr[2 * lane + 1] + b * inv;
  float ss2 = reduce32(ua * ua + ub * ub);
  float u0 = __shfl(ua, 0, 32);
  float inv2 = 1.f / sqrtf(fmaxf(fabsf(ss2 - 2.f * u0 * u0), LEPS));
  float* orow = OUT + (size_t)node * 64;
  orow[2 * lane] = ua * inv2;
  orow[2 * lane + 1] = ub * inv2;
}

extern "C" void kernel_launch(void* const* d_in, const int* in_sizes, int n_in,
                              void* d_out, int out_size, void* d_ws, size_t ws_size,
                              hipStream_t stream) {
  const float* x  = (const float*)d_in[0];
  const int*   ei = (const int*)d_in[1];   // [0,E)=src, [E,2E)=dst
  const float* ew = (const float*)d_in[2];
  const float* W1 = (const float*)d_in[3];
  const float* b1 = (const float*)d_in[4];
  const float* s1 = (const float*)d_in[5];
  const float* W2 = (const float*)d_in[6];
  const float* b2 = (const float*)d_in[7];
  const float* s2 = (const float*)d_in[8];

  int N = in_sizes[0] / 63;
  int E = in_sizes[2];

  float* p     = (float*)d_ws;
  float* hlin  = p + (size_t)N * 64;
  float* accum = hlin + (size_t)N * 64;
  float* h1    = accum + (size_t)N * 64;
  float* out   = (float*)d_out;

  int nodeBlocks = (N + 7) / 8;
  int gemmBlocks = (N + 127) / 128;
  int zeroBlocks = (N * 64 + 255) / 256;
  long long total = (long long)E * 64;
  int scatBlocks = (int)((total + 255) / 256);

  // Layer 1
  expmap_kernel<<<nodeBlocks, 256, 0, stream>>>(x, p, N);
  lorentz_linear_kernel<<<gemmBlocks, 256, 0, stream>>>(p, W1, b1, s1, hlin, N, 0);
  zero_kernel<<<zeroBlocks, 256, 0, stream>>>(accum, N * 64);
  scatter_kernel<<<scatBlocks, 256, 0, stream>>>(hlin, ei, ei + E, ew, accum, (unsigned)E);
  normalize_kernel<<<nodeBlocks, 256, 0, stream>>>(accum, h1, N);
  // Layer 2 (ReLU on input)
  lorentz_linear_kernel<<<gemmBlocks, 256, 0, stream>>>(h1, W2, b2, s2, hlin, N, 1);
  zero_kernel<<<zeroBlocks, 256, 0, stream>>>(accum, N * 64);
  scatter_kernel<<<scatBlocks, 256, 0, stream>>>(hlin, ei, ei + E, ew, accum, (unsigned)E);
  final_kernel<<<nodeBlocks, 256, 0, stream>>>(accum, h1, out, N);
}